// CoherenceEnhancingDiffusionLayer_15427522527846
// MI455X (gfx1250) — compile-verified
//
#include <hip/hip_runtime.h>
#include <cstdint>

typedef float v2f __attribute__((ext_vector_type(2)));
typedef float v8f __attribute__((ext_vector_type(8)));

#define IMG_H 256
#define IMG_W 256
#define NBATCH 8
#define NPIX (IMG_H * IMG_W)

#define TILE 32          // output tile (32x32)
#define GH   40          // gprod rows  = TILE + 8
#define GW   48          // gprod cols  = TILE + 16
#define GS   52          // gprod LDS row stride (floats)
#define TS   52          // tmp   LDS row stride (floats)

__device__ __forceinline__ int refl(int i, int n) {
    i = (i < 0) ? -i : i;
    return (i >= n) ? (2 * n - 2 - i) : i;
}

__device__ __forceinline__ float uz(const float* __restrict__ u, int r, int c) {
    return (r >= 0 && r < IMG_H && c >= 0 && c < IMG_W) ? u[r * IMG_W + c] : 0.f;
}

__device__ __forceinline__ void gauss9(float* g) {
    // matches torchvision GaussianBlur: sigma=4, ksize=9, normalized
    float s = 0.f;
#pragma unroll
    for (int t = 0; t < 9; ++t) {
        float x = (float)(t - 4) * 0.25f;       // xs / sigma
        g[t] = __expf(-0.5f * x * x);
        s += g[t];
    }
    float inv = 1.f / s;
#pragma unroll
    for (int t = 0; t < 9; ++t) g[t] *= inv;
}

// ---------------- pre-pass: RGB -> luma ----------------
__global__ void ced_luma(const float* __restrict__ x, float* __restrict__ u) {
    int i = blockIdx.x * blockDim.x + threadIdx.x;
    if (i >= NBATCH * NPIX) return;
    int b = i / NPIX, p = i % NPIX;
    const float* xb = x + (size_t)b * 3 * NPIX;
    u[i] = 0.299f * xb[p] + 0.587f * xb[NPIX + p] + 0.114f * xb[2 * NPIX + p];
}

// ---------------- pre-pass: gauss_rho = blur(u0), reflect pad ----------------
__global__ void ced_rho(const float* __restrict__ u0, float* __restrict__ rho) {
    int i = blockIdx.x * blockDim.x + threadIdx.x;
    if (i >= NBATCH * NPIX) return;
    int b = i / NPIX, p = i % NPIX;
    int r = p / IMG_W, c = p % IMG_W;
    const float* ub = u0 + (size_t)b * NPIX;
    float g[9];
    gauss9(g);
    float acc = 0.f;
#pragma unroll
    for (int dy = 0; dy < 9; ++dy) {
        int rr = refl(r + dy - 4, IMG_H);
        float rowacc = 0.f;
#pragma unroll
        for (int dx = 0; dx < 9; ++dx) {
            int cc = refl(c + dx - 4, IMG_W);
            rowacc += g[dx] * ub[rr * IMG_W + cc];
        }
        acc += g[dy] * rowacc;
    }
    rho[i] = acc;
}

// ---------------- one diffusion step, fully fused per 32x32 tile ----------------
__global__ __launch_bounds__(128) void ced_step(const float* __restrict__ usrc,
                                                const float* __restrict__ rho,
                                                float* __restrict__ udst,
                                                const float* __restrict__ dtp,
                                                const float* __restrict__ kp) {
    __shared__ float s_gp[3][GH * GS];    // gxx, gxy, gyy over (TILE+8) x (TILE+16)
    __shared__ float s_tmp[3][TILE * TS]; // vertical-blur result, TILE x (TILE+16)
    __shared__ float s_rho[TILE * TILE];  // gauss_rho tile (async-staged)

    const int tid  = threadIdx.x;
    const int lane = tid & 31;
    const int wave = tid >> 5;
    const int half = lane >> 4;           // 0: lanes 0-15, 1: lanes 16-31
    const int nn   = lane & 15;

    const int tilesx  = IMG_W / TILE;                 // 8
    const int tilespb = tilesx * (IMG_H / TILE);      // 64
    const int b  = blockIdx.x / tilespb;
    const int t  = blockIdx.x % tilespb;
    const int R0 = (t / tilesx) * TILE;
    const int C0 = (t % tilesx) * TILE;

    const float* u  = usrc + (size_t)b * NPIX;
    const float* rb = rho  + (size_t)b * NPIX;
    float*       ud = udst + (size_t)b * NPIX;

    // ---- stage 0: async-stage rho tile into LDS (ASYNCcnt-tracked DMA path) ----
    // 32x32 floats = 256 x 16B chunks; 128 threads x 2 chunks, uniform EXEC.
    for (int i = tid; i < (TILE * TILE) / 4; i += 128) {
        int rr = i >> 3;            // row 0..31
        int c4 = i & 7;             // 16B chunk within row
        const float* gsrc = rb + (R0 + rr) * IMG_W + C0 + c4 * 4;
        uint32_t ldsoff = (uint32_t)(uintptr_t)(&s_rho[rr * TILE + c4 * 4]);
        asm volatile("global_load_async_to_lds_b128 %0, %1, off"
                     :: "v"(ldsoff), "v"(gsrc) : "memory");
    }

    // ---- per-lane constant band-matrix fragments for the 9-tap Gaussian ----
    // W[q][j] = g[q - j] for 0 <= q-j <= 8 (24x16 band). Same per-lane values
    // serve as the A-fragment of the vertical blur (Wv[i][q] = g[q - i]).
    float g[9];
    gauss9(g);
    float wf0[6], wf1[6];
#pragma unroll
    for (int k = 0; k < 6; ++k) {
        int d0 = 4 * k + 2 * half - nn;   // K-index minus row/col index
        int d1 = d0 + 1;
        wf0[k] = (d0 >= 0 && d0 <= 8) ? g[d0] : 0.f;
        wf1[k] = (d1 >= 0 && d1 <= 8) ? g[d1] : 0.f;
    }

    // ---- stage 1: Sobel (zero pad) + products at reflect-mapped coords ----
    // gprod LDS (gr,gc) <-> global (R0-4+gr, C0-8+gc), reflect-mapped into image.
    for (int i = tid; i < GH * GW; i += 128) {   // 1920 = 15*128, uniform EXEC
        int gr = i / GW, gc = i % GW;
        int rr = refl(R0 - 4 + gr, IMG_H);
        int cc = refl(C0 - 8 + gc, IMG_W);
        float a00 = uz(u, rr - 1, cc - 1), a01 = uz(u, rr - 1, cc), a02 = uz(u, rr - 1, cc + 1);
        float a10 = uz(u, rr,     cc - 1),                          a12 = uz(u, rr,     cc + 1);
        float a20 = uz(u, rr + 1, cc - 1), a21 = uz(u, rr + 1, cc), a22 = uz(u, rr + 1, cc + 1);
        float gx = (a02 - a00) + 2.f * (a12 - a10) + (a22 - a20);
        float gy = (a20 - a00) + 2.f * (a21 - a01) + (a22 - a02);
        s_gp[0][gr * GS + gc] = gx * gx;
        s_gp[1][gr * GS + gc] = gx * gy;
        s_gp[2][gr * GS + gc] = gy * gy;
    }

    asm volatile("s_wait_asynccnt 0x0" ::: "memory");  // rho tile landed
    __syncthreads();

    // ---- stage 2: vertical 9-tap blur via WMMA f32 16x16x4 ----
    // tmp[i][j] = sum_q Wv[i][q] * gprod[r0+q][c0+j], q in [0,24), banded.
    // 3 fields x (2x3) 16x16 subtiles = 18 jobs; wave-uniform loop, EXEC full.
    for (int job = wave; job < 18; job += 4) {
        int f  = job / 6;
        int rc = job % 6;
        int r0 = (rc / 3) * 16;
        int c0 = (rc % 3) * 16;
        const float* gp = s_gp[f];
        v8f acc = {};
#pragma unroll
        for (int k = 0; k < 6; ++k) {
            v2f a; a.x = wf0[k]; a.y = wf1[k];                 // A = Wv chunk (const)
            int row = r0 + 4 * k + 2 * half;                    // B rows v + 2*half
            v2f bb; bb.x = gp[row * GS + c0 + nn];
                    bb.y = gp[(row + 1) * GS + c0 + nn];
            acc = __builtin_amdgcn_wmma_f32_16x16x4_f32(
                false, a, false, bb, (short)0, acc, false, false);
        }
        float* tp = s_tmp[f];
#pragma unroll
        for (int e = 0; e < 8; ++e)                             // C: row e + 8*half
            tp[(r0 + e + 8 * half) * TS + c0 + nn] = acc[e];
    }
    __syncthreads();

    // ---- stage 3: horizontal blur via WMMA + fused eigen/exp update ----
    // out[i][j] = sum_q tmp[r0+i][c0+4+q] * Wh[q][j]; one 16x16 subtile per wave.
    const float dt = *dtp;
    const float kv = *kp;
    const float inv_k2 = 1.f / (kv * kv);
    {
        int r0 = (wave >> 1) * 16;
        int c0 = (wave & 1) * 16;
        v8f S[3];
#pragma unroll
        for (int f = 0; f < 3; ++f) {
            const float* tp = s_tmp[f];
            v8f acc = {};
#pragma unroll
            for (int k = 0; k < 6; ++k) {
                int colb = c0 + 4 + 4 * k + 2 * half;           // A cols v + 2*half
                const float* rowp = tp + (r0 + nn) * TS;
                v2f a; a.x = rowp[colb]; a.y = rowp[colb + 1];
                v2f bb; bb.x = wf0[k]; bb.y = wf1[k];           // B = Wh chunk (const)
                acc = __builtin_amdgcn_wmma_f32_16x16x4_f32(
                    false, a, false, bb, (short)0, acc, false, false);
            }
            S[f] = acc;
        }
#pragma unroll
        for (int e = 0; e < 8; ++e) {
            int r = r0 + e + 8 * half;                           // row in tile
            int c = c0 + nn;
            float rhov = s_rho[r * TILE + c];
            float s11 = S[0][e] - rhov;
            float s12 = S[1][e] - rhov;
            float s22 = S[2][e] - rhov;
            float dd   = s11 - s22;
            float disc = __fsqrt_rn(dd * dd + 4.f * s12 * s12);
            float lam1 = (s11 + s22 + disc) * 0.5f;
            float lam2 = (s11 + s22 - disc) * 0.5f;
            float c1 = __expf(-(lam1 * lam1) * inv_k2);
            float c2 = __expf(-(lam2 * lam2) * inv_k2);
            int gidx = (R0 + r) * IMG_W + (C0 + c);
            ud[gidx] = u[gidx] + dt * (c1 * lam1 + c2 * lam2);
        }
    }
}

// ---------------- post-pass: repeat luma to 3 channels ----------------
__global__ void ced_repeat(const float* __restrict__ u, float* __restrict__ out) {
    int i = blockIdx.x * blockDim.x + threadIdx.x;
    if (i >= NBATCH * NPIX) return;
    int b = i / NPIX, p = i % NPIX;
    float v = u[i];
    float* ob = out + (size_t)b * 3 * NPIX;
    ob[p] = v; ob[NPIX + p] = v; ob[2 * NPIX + p] = v;
}

extern "C" void kernel_launch(void* const* d_in, const int* in_sizes, int n_in,
                              void* d_out, int out_size, void* d_ws, size_t ws_size,
                              hipStream_t stream) {
    const float* x   = (const float*)d_in[0];
    const float* dtp = (const float*)d_in[1];
    const float* kp  = (const float*)d_in[2];
    float* out = (float*)d_out;

    float* u0  = (float*)d_ws;
    float* u1  = u0 + (size_t)NBATCH * NPIX;
    float* rho = u1 + (size_t)NBATCH * NPIX;

    const int n = NBATCH * NPIX;
    ced_luma<<<(n + 255) / 256, 256, 0, stream>>>(x, u0);
    ced_rho<<<(n + 255) / 256, 256, 0, stream>>>(u0, rho);

    for (int it = 0; it < 100; ++it) {
        const float* src = (it & 1) ? u1 : u0;
        float*       dst = (it & 1) ? u0 : u1;
        ced_step<<<NBATCH * 64, 128, 0, stream>>>(src, rho, dst, dtp, kp);
    }
    // after 100 steps the result is back in u0
    ced_repeat<<<(n + 255) / 256, 256, 0, stream>>>(u0, out);
}